// ToRGBLayer_36507222016809
// MI455X (gfx1250) — compile-verified
//
#include <hip/hip_runtime.h>
#include <hip/hip_bf16.h>

typedef __attribute__((ext_vector_type(2))) float v2f;
typedef __attribute__((ext_vector_type(8))) float v8f;

#define HW_PIX   65536   // 256*256
#define C_IN_    128
#define W_DIM_   512
#define NPT      4       // 16-pixel tiles per wave

// ---------------------------------------------------------------------------
// Kernel 1: build Weff[b][m][i] = styles[b][i] * w2d[m][i]  (m padded to 16,
// rows 3..15 = 0).  Layout chosen so the WMMA A-matrix (16x4 f32) tile for
// channel group kk is a single b64 load per lane.
// ---------------------------------------------------------------------------
__global__ void style_kernel(const float* __restrict__ w,        // (16,512)
                             const float* __restrict__ weight,   // (3,128)
                             const float* __restrict__ affine_w, // (384,512)
                             const float* __restrict__ affine_b, // (384)
                             float* __restrict__ weff)           // (16,16,128)
{
    __shared__ float wsh[W_DIM_];
    const int b = blockIdx.x;
    const int i = threadIdx.x;                  // 0..127 -> input channel

    for (int k = i; k < W_DIM_; k += 128) wsh[k] = w[b * W_DIM_ + k];
    __syncthreads();

    const float fc_gain = 0.04419417382415922f;   // 1/sqrt(512)
    float a0 = 0.f, a1 = 0.f, a2 = 0.f;
    const float* r0 = affine_w + (size_t)(i        ) * W_DIM_;
    const float* r1 = affine_w + (size_t)(i + 128  ) * W_DIM_;
    const float* r2 = affine_w + (size_t)(i + 256  ) * W_DIM_;
    #pragma unroll 4
    for (int k = 0; k < W_DIM_; ++k) {
        const float wv = wsh[k];
        a0 = fmaf(wv, r0[k], a0);
        a1 = fmaf(wv, r1[k], a1);
        a2 = fmaf(wv, r2[k], a2);
    }
    const float s1 = a0 * fc_gain + affine_b[i];
    const float s2 = a1 * fc_gain + affine_b[i + 128];
    const float s3 = a2 * fc_gain + affine_b[i + 256];
    const float style = (s1 * s2 + s3) * 0.08838834764831845f; // 1/sqrt(128)

    float* wb = weff + (size_t)b * 16 * C_IN_;
    #pragma unroll
    for (int o = 0; o < 16; ++o) {
        const float v = (o < 3) ? style * weight[o * C_IN_ + i] : 0.0f;
        wb[o * C_IN_ + i] = v;
    }
}

// ---------------------------------------------------------------------------
// Kernel 2: out[b][m][p] = sum_i Weff[b][m][i] * x[b][i][p], bias, clamp.
// One wave = NPT adjacent 16-pixel tiles of one batch image.
// D(16x16) = A(16x4: Weff tile) x B(4x16: x tile), accumulated over 32 K-groups.
// ---------------------------------------------------------------------------
__global__ void __launch_bounds__(256)
torgb_wmma_kernel(const float* __restrict__ x,    // (16,128,65536)
                  const float* __restrict__ weff, // (16,16,128)
                  const float* __restrict__ bias, // (3)
                  float* __restrict__ out)        // (16,3,65536)
{
    const int lane  = threadIdx.x & 31;
    const int wave  = threadIdx.x >> 5;
    const long long gwave = (long long)blockIdx.x * (blockDim.x >> 5) + wave;
    const long long tile0 = gwave * NPT;                // first 16-pixel tile
    const int tiles_per_batch = HW_PIX / 16;            // 4096 (divisible by NPT)
    const int b  = (int)(tile0 / tiles_per_batch);
    const int p0 = (int)(tile0 % tiles_per_batch) * 16; // first pixel index

    const float* xb = x    + (size_t)b * C_IN_ * HW_PIX;
    const float* wb = weff + (size_t)b * 16 * C_IN_;

    const int m  = lane & 15;      // A row (out-channel) / B column (pixel)
    const int kh = lane >> 4;      // K half-select per ISA A/B f32 layouts

    // A tile base for this lane: floats {Weff[m][4kk+2kh], Weff[m][4kk+2kh+1]}
    const v2f* aptr = (const v2f*)(wb + m * C_IN_ + 2 * kh);

    v8f acc[NPT];
    #pragma unroll
    for (int j = 0; j < NPT; ++j) acc[j] = (v8f){0,0,0,0,0,0,0,0};

    for (int kk = 0; kk < 32; ++kk) {
        const v2f a = aptr[2 * kk];  // float offset 4*kk -> b64 load (L2-hot)
        // B tile: lanes 0-15 hold channels (4kk,4kk+1), lanes 16-31 (4kk+2,4kk+3)
        const float* xc = xb + (size_t)(4 * kk + 2 * kh) * HW_PIX + p0 + m;
        #pragma unroll
        for (int j = 0; j < NPT; ++j) {
            v2f bv;
            bv.x = __builtin_nontemporal_load(xc + j * 16);           // channel c
            bv.y = __builtin_nontemporal_load(xc + j * 16 + HW_PIX);  // channel c+1
            // 8 args: (neg_a, A, neg_b, B, c_mod, C, reuse_a, reuse_b)
            acc[j] = __builtin_amdgcn_wmma_f32_16x16x4_f32(
                false, a, false, bv, (short)0, acc[j], false, false);
        }
    }

    // Epilogue: rows M=0..2 live in acc[j][0..2] on lanes 0-15 (N = lane).
    if (lane < 16) {
        const float b0 = bias[0], b1 = bias[1], b2 = bias[2];
        float* ob = out + (size_t)b * 3 * HW_PIX + p0 + lane;
        #pragma unroll
        for (int j = 0; j < NPT; ++j) {
            float v0 = fminf(fmaxf(acc[j][0] + b0, -256.f), 256.f);
            float v1 = fminf(fmaxf(acc[j][1] + b1, -256.f), 256.f);
            float v2 = fminf(fmaxf(acc[j][2] + b2, -256.f), 256.f);
            __builtin_nontemporal_store(v0, ob + j * 16);
            __builtin_nontemporal_store(v1, ob + j * 16 + HW_PIX);
            __builtin_nontemporal_store(v2, ob + j * 16 + 2 * HW_PIX);
        }
    }
}

extern "C" void kernel_launch(void* const* d_in, const int* in_sizes, int n_in,
                              void* d_out, int out_size, void* d_ws, size_t ws_size,
                              hipStream_t stream) {
    const float* x        = (const float*)d_in[0]; // (16,128,256,256)
    const float* w        = (const float*)d_in[1]; // (16,512)
    const float* weight   = (const float*)d_in[2]; // (3,128,1,1)
    const float* bias     = (const float*)d_in[3]; // (3)
    const float* affine_w = (const float*)d_in[4]; // (384,512)
    const float* affine_b = (const float*)d_in[5]; // (384)
    float* out  = (float*)d_out;                   // (16,3,256,256)
    float* weff = (float*)d_ws;                    // 16*16*128 floats = 128 KB

    style_kernel<<<16, 128, 0, stream>>>(w, weight, affine_w, affine_b, weff);

    // 16 batches * 4096 tiles / (8 waves * NPT tiles) = 2048 blocks
    torgb_wmma_kernel<<<2048, 256, 0, stream>>>(x, weff, bias, out);
}